// GlobalSLC_28647431864612
// MI455X (gfx1250) — compile-verified
//
#include <hip/hip_runtime.h>

// ---------------------------------------------------------------------------
// GlobalSLC on MI455X (gfx1250), fp32 throughout (matches reference dtype).
//
// Static branch:   3x HBM-streaming GEMM  P = ws @ y  using V_WMMA_F32_16X16X4_F32,
//                  ws tiles staged with GLOBAL_LOAD_ASYNC_TO_LDS_B128 when available.
// Dynamic branch:  factorized, wd never materialized:
//                  wd @ z = (x@wp) @ (x^T z)   -> 32x32 reductions + skinny GEMMs
// ---------------------------------------------------------------------------

#define NN   8192
#define CC   32

typedef float v2f __attribute__((ext_vector_type(2)));
typedef float v8f __attribute__((ext_vector_type(8)));

#if defined(__AMDGCN__) && __has_builtin(__builtin_amdgcn_global_load_async_to_lds_b128)
#define HAVE_ASYNC_LDS 1
typedef int v4i __attribute__((vector_size(4 * sizeof(int))));
typedef __attribute__((address_space(1))) v4i as1_v4i;   // global (HIP __device__)
typedef __attribute__((address_space(3))) v4i as3_v4i;   // LDS
#endif

__device__ __forceinline__ void wait_async_done() {
#if defined(HAVE_ASYNC_LDS)
#if __has_builtin(__builtin_amdgcn_s_wait_asynccnt)
  __builtin_amdgcn_s_wait_asynccnt(0);
#else
  asm volatile("s_wait_asynccnt 0" ::: "memory");
#endif
#endif
}

// ---------------- big GEMM: partial = ws[rows, ksplit] @ yin ----------------
// 64 rows/block, 2 waves (64 thr), each wave a 32x32 output slab.
// K split in two halves across blockIdx parity -> 256 WGs for occupancy.
static constexpr int BR  = 64;    // rows per block
static constexpr int KC  = 128;   // k-chunk staged in LDS
static constexpr int LDW = 132;   // ws LDS row stride (floats): conflict-free A reads

// x tile packed layout: xt[(k>>2)*128 + col*4 + (k&3)]
//   -> B fragment (k, k+1 at col) is one aligned b64, banks 4*col + 2*half (conflict-free)

__global__ __launch_bounds__(64)
void kbig(const float* __restrict__ ws, const float* __restrict__ yin,
          float* __restrict__ part) {
  __shared__ float lw[BR * LDW];      // 33.8 KB
  __shared__ float lxp[KC * CC];      // 16 KB, packed k-groups
  const int t    = threadIdx.x;
  const int lane = t & 31;
  const int wave = t >> 5;               // 0..1
  const int bid  = blockIdx.x;
  const int rb   = (bid >> 1) * BR;      // row block
  const int ks   = bid & 1;              // k split
  const int k0   = ks * (NN / 2);
  const int half = lane >> 4;            // lane-half selects K pair (ISA 7.12.2)
  const int l16  = lane & 15;
  const int rowA = wave * 32;            // wave's row base inside block

  v8f acc00 = {}; v8f acc01 = {}; v8f acc10 = {}; v8f acc11 = {};

  for (int kb = k0; kb < k0 + NN / 2; kb += KC) {
    // ---- stage ws tile: 64 rows x 128 k ----
    {
      const int c4 = (t & 31) * 4;                   // 0..124
#if defined(HAVE_ASYNC_LDS)
      #pragma unroll 8
      for (int i = 0; i < 32; ++i) {
        const int r = (t >> 5) + 2 * i;              // 0..63
        __builtin_amdgcn_global_load_async_to_lds_b128(
            (as1_v4i*)(ws + (size_t)(rb + r) * NN + kb + c4),
            (as3_v4i*)(&lw[r * LDW + c4]), 0, 0);
      }
#else
      #pragma unroll 4
      for (int i = 0; i < 32; ++i) {
        const int r = (t >> 5) + 2 * i;              // 0..63
        const float4 v = *(const float4*)(ws + (size_t)(rb + r) * NN + kb + c4);
        *(float4*)(&lw[r * LDW + c4]) = v;
      }
#endif
    }
    // ---- stage yin tile: 128 k x 32 cols, repacked into k-groups ----
    {
      const int c4 = (t & 7) * 4;                    // 0..28
      #pragma unroll 4
      for (int i = 0; i < 16; ++i) {
        const int k = (t >> 3) + 8 * i;              // 0..127
        const float4 v = *(const float4*)(yin + (size_t)(kb + k) * CC + c4);
        const int pb = (k >> 2) * 128 + (k & 3);
        lxp[pb + (c4 + 0) * 4] = v.x;
        lxp[pb + (c4 + 1) * 4] = v.y;
        lxp[pb + (c4 + 2) * 4] = v.z;
        lxp[pb + (c4 + 3) * 4] = v.w;
      }
    }
    wait_async_done();
    __syncthreads();

    // ---- 32 k-steps x 4 WMMA ----
    #pragma unroll 4
    for (int kk = 0; kk < KC; kk += 4) {
      const int kf = kk + 2 * half;                  // K index pair per lane-half
      const v2f aLo = *(const v2f*)(&lw[(rowA + l16) * LDW + kf]);
      const v2f aHi = *(const v2f*)(&lw[(rowA + 16 + l16) * LDW + kf]);
      const int xb = (kk >> 2) * 128 + 2 * half;
      const v2f b0 = *(const v2f*)(&lxp[xb + l16 * 4]);
      const v2f b1 = *(const v2f*)(&lxp[xb + (16 + l16) * 4]);
      acc00 = __builtin_amdgcn_wmma_f32_16x16x4_f32(false, aLo, false, b0, (short)0, acc00, false, false);
      acc01 = __builtin_amdgcn_wmma_f32_16x16x4_f32(false, aLo, false, b1, (short)0, acc01, false, false);
      acc10 = __builtin_amdgcn_wmma_f32_16x16x4_f32(false, aHi, false, b0, (short)0, acc10, false, false);
      acc11 = __builtin_amdgcn_wmma_f32_16x16x4_f32(false, aHi, false, b1, (short)0, acc11, false, false);
    }
    __syncthreads();
  }

  // ---- store partial tiles (C/D layout: M = j + 8*half, N = l16) ----
  float* P = part + (size_t)ks * NN * CC;
  #pragma unroll
  for (int j = 0; j < 8; ++j) {
    const int m0 = rb + rowA + j + 8 * half;
    const int m1 = m0 + 16;
    P[(size_t)m0 * CC + l16]      = acc00[j];
    P[(size_t)m0 * CC + 16 + l16] = acc01[j];
    P[(size_t)m1 * CC + l16]      = acc10[j];
    P[(size_t)m1 * CC + 16 + l16] = acc11[j];
  }
}

// ------------- combine K-split partials + Chebyshev axpby -------------------
// Y = alpha*(P0 + P1) + beta*Sub
__global__ __launch_bounds__(256)
void kcomb(const float* __restrict__ part, const float* __restrict__ sub,
           float alpha, float beta, float* __restrict__ Y) {
  const size_t i = (size_t)blockIdx.x * 256 + threadIdx.x;   // float4 index
  const float4 a = ((const float4*)part)[i];
  const float4 b = ((const float4*)(part + (size_t)NN * CC))[i];
  const float4 s = ((const float4*)sub)[i];
  float4 r;
  r.x = alpha * (a.x + b.x) + beta * s.x;
  r.y = alpha * (a.y + b.y) + beta * s.y;
  r.z = alpha * (a.z + b.z) + beta * s.z;
  r.w = alpha * (a.w + b.w) + beta * s.w;
  ((float4*)Y)[i] = r;
}

// ------------- G = A^T @ B  (32x32), deterministic two-stage ---------------
__global__ __launch_bounds__(256)
void kred1(const float* __restrict__ A, const float* __restrict__ B,
           float* __restrict__ gpart) {
  const int t = threadIdx.x, b = blockIdx.x;     // 64 blocks
  const int j = t & 31, i0 = t >> 5;             // i0 in 0..7
  float a0 = 0.f, a1 = 0.f, a2 = 0.f, a3 = 0.f;
  const int r0 = b * (NN / 64);
  for (int r = r0; r < r0 + NN / 64; ++r) {
    const float bv = B[(size_t)r * CC + j];
    a0 += A[(size_t)r * CC + i0]      * bv;
    a1 += A[(size_t)r * CC + i0 + 8]  * bv;
    a2 += A[(size_t)r * CC + i0 + 16] * bv;
    a3 += A[(size_t)r * CC + i0 + 24] * bv;
  }
  float* g = gpart + (size_t)b * 1024;
  g[(i0)      * 32 + j] = a0;
  g[(i0 + 8)  * 32 + j] = a1;
  g[(i0 + 16) * 32 + j] = a2;
  g[(i0 + 24) * 32 + j] = a3;
}

__global__ __launch_bounds__(256)
void kred2(const float* __restrict__ gpart, float* __restrict__ G) {
  const int t = threadIdx.x;
  for (int e = t; e < 1024; e += 256) {
    float s = 0.f;
    for (int p = 0; p < 64; ++p) s += gpart[(size_t)p * 1024 + e];
    G[e] = s;
  }
}

// ------------- Z = alpha*(A @ G) + beta*Sub   (A: Nx32, G: 32x32) ----------
__global__ __launch_bounds__(256)
void ksmall(const float* __restrict__ A, const float* __restrict__ G,
            const float* __restrict__ sub, float alpha, float beta,
            float* __restrict__ Z) {
  __shared__ float g[1024];
  const int t = threadIdx.x;
  for (int e = t; e < 1024; e += 256) g[e] = G[e];
  __syncthreads();
  const int c = t & 31;
  const size_t r = (size_t)blockIdx.x * 8 + (t >> 5);
  float acc = 0.f;
  #pragma unroll
  for (int k = 0; k < 32; ++k) acc += A[r * CC + k] * g[k * 32 + c];
  Z[r * CC + c] = alpha * acc + beta * sub[r * CC + c];
}

// ------------- out = 2*x*ts0 + y1*ts1 + y2*ts2 + y3*ts3 + z*td -------------
__global__ __launch_bounds__(256)
void kfinal(const float* __restrict__ x,  const float* __restrict__ y1,
            const float* __restrict__ y2, const float* __restrict__ y3,
            const float* __restrict__ z1, const float* __restrict__ z2,
            const float* __restrict__ z3, const float* __restrict__ ts,
            const float* __restrict__ td, float* __restrict__ out) {
  __shared__ float w[7 * 1024];
  const int t = threadIdx.x;
  for (int e = t; e < 1024; e += 256) {
    w[e]        = 2.0f * ts[e];        // shared k=0 term appears in both branches
    w[1024 + e] = ts[1024 + e];
    w[2048 + e] = ts[2048 + e];
    w[3072 + e] = ts[3072 + e];
    w[4096 + e] = td[1024 + e];
    w[5120 + e] = td[2048 + e];
    w[6144 + e] = td[3072 + e];
  }
  __syncthreads();
  const int c = t & 31;
  const size_t r = (size_t)blockIdx.x * 8 + (t >> 5);
  float acc = 0.f;
  #pragma unroll
  for (int k = 0; k < 32; ++k) {
    acc += x [r * CC + k] * w[k * 32 + c];
    acc += y1[r * CC + k] * w[1024 + k * 32 + c];
    acc += y2[r * CC + k] * w[2048 + k * 32 + c];
    acc += y3[r * CC + k] * w[3072 + k * 32 + c];
    acc += z1[r * CC + k] * w[4096 + k * 32 + c];
    acc += z2[r * CC + k] * w[5120 + k * 32 + c];
    acc += z3[r * CC + k] * w[6144 + k * 32 + c];
  }
  out[r * CC + c] = acc;
}

// ---------------------------------------------------------------------------
extern "C" void kernel_launch(void* const* d_in, const int* in_sizes, int n_in,
                              void* d_out, int out_size, void* d_ws, size_t ws_size,
                              hipStream_t stream) {
  const float* x  = (const float*)d_in[0];   // 8192 x 32
  const float* ws = (const float*)d_in[1];   // 8192 x 8192
  const float* wp = (const float*)d_in[2];   // 32 x 32
  const float* ts = (const float*)d_in[3];   // 4 x 32 x 32
  const float* td = (const float*)d_in[4];   // 4 x 32 x 32
  float* out = (float*)d_out;

  float* S = (float*)d_ws;
  const size_t M = (size_t)NN * CC;          // 262144 floats
  float* xp    = S;                          // x @ wp
  float* y1    = S + 1 * M;
  float* y2    = S + 2 * M;
  float* y3    = S + 3 * M;
  float* z1    = S + 4 * M;
  float* z2    = S + 5 * M;
  float* z3    = S + 6 * M;
  float* P     = S + 7 * M;                  // 2 K-split partials (2*M)
  float* gpart = S + 9 * M;                  // 64 x 1024
  float* G     = gpart + 64 * 1024;          // 32 x 32

  // ---------------- dynamic branch (wd factorized, ~free) ----------------
  kred1 <<<64,   256, 0, stream>>>(x, x, gpart);            // G = x^T x
  kred2 <<<1,    256, 0, stream>>>(gpart, G);
  ksmall<<<1024, 256, 0, stream>>>(x,  wp, x, 1.f,  0.f, xp);   // xp = x @ wp
  ksmall<<<1024, 256, 0, stream>>>(xp, G,  x, 1.f,  0.f, z1);   // z1 = wd @ x
  kred1 <<<64,   256, 0, stream>>>(x, z1, gpart);           // H = x^T z1
  kred2 <<<1,    256, 0, stream>>>(gpart, G);
  ksmall<<<1024, 256, 0, stream>>>(xp, G,  x, 2.f, -1.f, z2);   // z2 = 2 wd z1 - x
  kred1 <<<64,   256, 0, stream>>>(x, z2, gpart);           // H = x^T z2
  kred2 <<<1,    256, 0, stream>>>(gpart, G);
  ksmall<<<1024, 256, 0, stream>>>(xp, G, z1, 2.f, -1.f, z3);   // z3 = 2 wd z2 - z1

  // ---------------- static branch (3 HBM-streaming WMMA passes) ----------
  kbig <<<256, 64,  0, stream>>>(ws, x,  P);
  kcomb<<<256, 256, 0, stream>>>(P, x,  1.f,  0.f, y1);     // y1 = ws x
  kbig <<<256, 64,  0, stream>>>(ws, y1, P);
  kcomb<<<256, 256, 0, stream>>>(P, x,  2.f, -1.f, y2);     // y2 = 2 ws y1 - x
  kbig <<<256, 64,  0, stream>>>(ws, y2, P);
  kcomb<<<256, 256, 0, stream>>>(P, y1, 2.f, -1.f, y3);     // y3 = 2 ws y2 - y1

  // ---------------- fused output --------------------------------------
  kfinal<<<1024, 256, 0, stream>>>(x, y1, y2, y3, z1, z2, z3, ts, td, out);
}